// DirectDepthMapper_69887707841068
// MI455X (gfx1250) — compile-verified
//
#include <hip/hip_runtime.h>
#include <math.h>

// DirectDepthMapper for MI455X (gfx1250, wave32).
// zero -> accumulate (float4 loads, WMMA f32 16x16x4 pose transform, wave
// match-any aggregated u32 atomics) -> finalize (gate + u32->f32 in place).
// All mask logic is bitwise (branchless) to avoid exec-mask control flow.

typedef __attribute__((ext_vector_type(2))) float v2f;
typedef __attribute__((ext_vector_type(8))) float v8f;

__device__ __forceinline__ unsigned ballot32(int p) {
  return __builtin_amdgcn_ballot_w32(p != 0);
}

__global__ __launch_bounds__(256) void ddm_zero(unsigned* __restrict__ hist,
                                                unsigned* __restrict__ counters,
                                                int nbins) {
  const int i = blockIdx.x * blockDim.x + threadIdx.x;
  if (i < nbins) hist[i] = 0u;
  if (i < 2) counters[i] = 0u;
}

__global__ __launch_bounds__(256) void ddm_main(const float* __restrict__ depth,
                                                const float* __restrict__ pose,
                                                unsigned* __restrict__ hist,
                                                unsigned* __restrict__ counters,
                                                int w, int wshift, int n, int M) {
  __shared__ unsigned s_n1, s_n2;
  if (threadIdx.x == 0) { s_n1 = 0u; s_n2 = 0u; }
  __syncthreads();

  const int tid = blockIdx.x * blockDim.x + threadIdx.x;
  const int live = (int)((tid * 4) < n);      // tail threads: masks forced off
  int base = tid * 4;
  if (base > n - 4) base = n - 4;             // clamp: loads stay in bounds,
                                              // EXEC stays full for WMMA
  const int lane = threadIdx.x & 31;

  // 16B vectorized depth load (w is a multiple of 4 -> same row).
  const float4 d4 = *(const float4*)(depth + base);
  const float dv[4] = {d4.x, d4.y, d4.z, d4.w};

  const float fx = 0.5f * (float)w;
  const float rf = 1.0f / fx;                 // exact: fx is a power of two
  const float cxy = (float)((int)fx - 1);     // cx == cy (square image)

  const int xb = base & (w - 1);              // x of first pixel in the quad
  const float yf = (float)(base >> wshift);   // y (same for all 4)
  const float pyc = (yf - cxy) * rf;          // py = d * pyc

  // A (MxK = 16x4) for V_WMMA_F32_16X16X4_F32: row M = pose row (M & 3);
  // lanes 0-15 hold K={0,1} in vgpr{0,1}, lanes 16-31 hold K={2,3}.
  const int arow = lane & 3;
  const int kbase = (lane < 16) ? 0 : 2;
  v2f a;
  a[0] = pose[arow * 4 + kbase];
  a[1] = pose[arow * 4 + kbase + 1];

  const int j0 = lane & 15;
  const bool lo = (lane < 16);
  unsigned n1acc = 0u, n2acc = 0u;

#pragma unroll
  for (int q = 0; q < 4; ++q) {
    const float d = dv[q];
    const float px = d * ((float)(xb + q) - cxy) * rf;
    const float py = d * pyc;
    const float pz = d;

    // B (KxN = 4x16): vgpr0 = row K0/px (lanes 0-15), K2/py (lanes 16-31);
    //                 vgpr1 = row K1/pz, K3/1.  Group0 = points of lanes 0-15,
    //                 group1 = points of lanes 16-31.
    const float px0 = __shfl(px, j0, 32);
    const float py0 = __shfl(py, j0, 32);
    const float pz0 = __shfl(pz, j0, 32);
    const float px1 = __shfl(px, 16 + j0, 32);
    const float py1 = __shfl(py, 16 + j0, 32);
    const float pz1 = __shfl(pz, 16 + j0, 32);

    v2f b0, b1;
    b0[0] = lo ? px0 : py0;
    b0[1] = lo ? pz0 : 1.0f;
    b1[0] = lo ? px1 : py1;
    b1[1] = lo ? pz1 : 1.0f;

    v8f cz = {};
    const v8f g0 = __builtin_amdgcn_wmma_f32_16x16x4_f32(false, a, false, b0,
                                                         (short)0, cz, false, false);
    const v8f g1 = __builtin_amdgcn_wmma_f32_16x16x4_f32(false, a, false, b1,
                                                         (short)0, cz, false, false);

    // D: lane L holds D[M][N=L&15], M = vgpr + 8*(L>>4); A rows repeat mod 4,
    // so rows {8,9,10} == {0,1,2}: each lane reads its own point's gx,gy',gz.
    const float gx  = lo ? g0[0] : g1[0];
    const float gyr = lo ? g0[1] : g1[1];
    const float gz  = lo ? g0[2] : g1[2];
    const float gy = -gyr;  // + CAMERA_HEIGHT (0)

    // Branchless predicate logic (bitwise, no short-circuit control flow).
    const float apz = fabsf(pz);
    const int mask1 = (int)(apz < 4.0f) & (int)(apz >= 0.1f) & live;
    const int mask2 = mask1 & (int)(gy > 0.0f) & (int)(gy < 1.0f);

    const int r  = __float2int_rn(gz * 10.0f + 200.0f);  // RNE == jnp.round
    const int cc = __float2int_rn(gx * 10.0f + 200.0f);
    const int inb = (int)(((unsigned)r < (unsigned)M) &
                          ((unsigned)cc < (unsigned)M));
    const int valid = mask2 & inb;
    const int bin = r * M + cc;

    n1acc += (unsigned)__popc(ballot32(mask1));
    n2acc += (unsigned)__popc(ballot32(mask2));

    // Match-any wave aggregation: one global atomic per distinct bin per wave.
    unsigned active = ballot32(valid);
    while (active) {
      const int leader = __ffs(active) - 1;
      const int lbin = __shfl(bin, leader, 32);
      const unsigned match = ballot32(valid & (int)(bin == lbin));
      if (lane == leader) atomicAdd(&hist[lbin], (unsigned)__popc(match));
      active &= ~match;
    }
  }

  if (lane == 0) {
    if (n1acc) atomicAdd(&s_n1, n1acc);
    if (n2acc) atomicAdd(&s_n2, n2acc);
  }
  __syncthreads();
  if (threadIdx.x == 0) {
    if (s_n1) atomicAdd(&counters[0], s_n1);
    if (s_n2) atomicAdd(&counters[1], s_n2);
  }
}

__global__ __launch_bounds__(256) void ddm_finalize(const unsigned* __restrict__ counters,
                                                    float* __restrict__ out,
                                                    int nbins) {
  const int i = blockIdx.x * blockDim.x + threadIdx.x;
  if (i >= nbins) return;
  const bool ok = (counters[0] >= 20u) && (counters[1] > 10u);
  const unsigned cnt = ((const unsigned*)out)[i];  // in-place u32 -> f32
  out[i] = ok ? (float)cnt : 0.0f;
}

extern "C" void kernel_launch(void* const* d_in, const int* in_sizes, int n_in,
                              void* d_out, int out_size, void* d_ws, size_t ws_size,
                              hipStream_t stream) {
  const float* depth = (const float*)d_in[0];
  const float* pose  = (const float*)d_in[1];

  const int n = in_sizes[0];                          // 2048*2048
  const int w = (int)(sqrt((double)n) + 0.5);         // 2048 (power of two)
  int wshift = 0;
  while ((1 << (wshift + 1)) <= w) ++wshift;          // log2(w) = 11
  const int M = (int)(sqrt((double)out_size) + 0.5);  // 400

  unsigned* histu = (unsigned*)d_out;
  float* outf = (float*)d_out;
  unsigned* counters = (unsigned*)d_ws;               // 2 x u32

  const int zb = (out_size + 255) / 256;              // 625 blocks
  ddm_zero<<<zb, 256, 0, stream>>>(histu, counters, out_size);

  const int nthreads = (n + 3) / 4;                   // 4 pixels per thread
  const int mb = (nthreads + 255) / 256;              // 4096 blocks
  ddm_main<<<mb, 256, 0, stream>>>(depth, pose, histu, counters, w, wshift, n, M);

  ddm_finalize<<<zb, 256, 0, stream>>>(counters, outf, out_size);
}